// VolumeRenderer_46145128628686
// MI455X (gfx1250) — compile-verified
//
#include <hip/hip_runtime.h>
#include <hip/hip_bf16.h>

// Volume renderer: N rays x P=128 samples.
// One wave32 per ray; lane l owns contiguous samples [4l, 4l+3].
// Memory-bound streaming kernel (~168 MB in -> ~7us at 23.3 TB/s).

#define VR_P 128
#define TERMINAL_DELTA 1e10f

__global__ __launch_bounds__(256) void VolumeRenderer_kernel(
    const float* __restrict__ depth_values,  // [N, 128]
    const float* __restrict__ density,       // [N, 128]
    const float* __restrict__ feature,       // [N, 128, 3]
    float* __restrict__ out_feat,            // [N, 3]
    float* __restrict__ out_depth,           // [N]
    int N)
{
    const int wavesPerBlock = blockDim.x >> 5;
    const int ray  = blockIdx.x * wavesPerBlock + (threadIdx.x >> 5);
    const int lane = threadIdx.x & 31;
    if (ray >= N) return;  // whole wave exits together (one ray per wave)

    const size_t base = (size_t)ray * VR_P + (size_t)lane * 4;  // sample index of lane's first sample
    const float* fptr = feature + base * 3;

    // gfx1250-specific prefetch path (global_prefetch_b8) on the widest stream.
    __builtin_prefetch(fptr, 0, 0);

    // ---- coalesced B128 loads ----
    const float4 d = *reinterpret_cast<const float4*>(depth_values + base);
    const float4 s = *reinterpret_cast<const float4*>(density + base);
    const float4 q0 = *reinterpret_cast<const float4*>(fptr);       // s0.xyz, s1.x
    const float4 q1 = *reinterpret_cast<const float4*>(fptr + 4);   // s1.yz, s2.xy
    const float4 q2 = *reinterpret_cast<const float4*>(fptr + 8);   // s2.z, s3.xyz

    // ---- deltas (next lane's first depth supplies the 4th delta) ----
    float nextd = __shfl_down(d.x, 1, 32);
    float del0 = d.y - d.x;
    float del1 = d.z - d.y;
    float del2 = d.w - d.z;
    float del3 = (lane == 31) ? TERMINAL_DELTA : (nextd - d.w);

    // ---- per-segment transmittance att = exp(-delta*density) ----
    float a0 = __expf(-del0 * s.x);
    float a1 = __expf(-del1 * s.y);
    float a2 = __expf(-del2 * s.z);
    float a3 = __expf(-del3 * s.w);

    // ---- exclusive cumprod across the ray ----
    // lane-local product, then multiplicative inclusive scan over 32 lanes.
    float p = a0 * a1 * a2 * a3;
    float inc = p;
    #pragma unroll
    for (int off = 1; off < 32; off <<= 1) {
        float up = __shfl_up(inc, off, 32);
        if (lane >= off) inc *= up;
    }
    float Tb = __shfl_up(inc, 1, 32);   // exclusive: product of all lanes < this one
    if (lane == 0) Tb = 1.0f;

    float T0 = Tb;
    float T1 = T0 * a0;
    float T2 = T1 * a1;
    float T3 = T2 * a2;

    float w0 = T0 * (1.0f - a0);
    float w1 = T1 * (1.0f - a1);
    float w2 = T2 * (1.0f - a2);
    float w3 = T3 * (1.0f - a3);

    // ---- weighted feature accumulation (channel c of sample k is f[3k+c]) ----
    float fx = w0 * q0.x + w1 * q0.w + w2 * q1.z + w3 * q2.y;
    float fy = w0 * q0.y + w1 * q1.x + w2 * q1.w + w3 * q2.z;
    float fz = w0 * q0.z + w1 * q1.y + w2 * q2.x + w3 * q2.w;

    // ---- first sample with density > 0 ----
    int cand = 1 << 30;
    if      (s.x > 0.0f) cand = lane * 4 + 0;
    else if (s.y > 0.0f) cand = lane * 4 + 1;
    else if (s.z > 0.0f) cand = lane * 4 + 2;
    else if (s.w > 0.0f) cand = lane * 4 + 3;

    // ---- butterfly reductions (sum for feat, min for first-hit index) ----
    #pragma unroll
    for (int off = 16; off > 0; off >>= 1) {
        fx += __shfl_xor(fx, off, 32);
        fy += __shfl_xor(fy, off, 32);
        fz += __shfl_xor(fz, off, 32);
        int oc = __shfl_xor(cand, off, 32);
        cand = (oc < cand) ? oc : cand;
    }

    if (lane == 0) {
        out_feat[(size_t)ray * 3 + 0] = fx;
        out_feat[(size_t)ray * 3 + 1] = fy;
        out_feat[(size_t)ray * 3 + 2] = fz;
        out_depth[ray] = (cand < VR_P) ? (1.0f - (float)cand * (1.0f / 128.0f)) : 0.0f;
    }
}

extern "C" void kernel_launch(void* const* d_in, const int* in_sizes, int n_in,
                              void* d_out, int out_size, void* d_ws, size_t ws_size,
                              hipStream_t stream) {
    (void)n_in; (void)d_ws; (void)ws_size; (void)out_size;
    const float* depth_values = (const float*)d_in[0];
    const float* density      = (const float*)d_in[1];
    const float* feature      = (const float*)d_in[2];

    const int N = in_sizes[1] / VR_P;   // density is [N, 128]

    float* out       = (float*)d_out;
    float* out_feat  = out;                   // [N,3] first
    float* out_depth = out + (size_t)N * 3;   // then [N,1]

    const int threads = 256;                  // 8 waves = 8 rays per block
    const int raysPerBlock = threads / 32;
    const int blocks = (N + raysPerBlock - 1) / raysPerBlock;

    VolumeRenderer_kernel<<<blocks, threads, 0, stream>>>(
        depth_values, density, feature, out_feat, out_depth, N);
}